// Spiking1DLIFLayer_86019605004462
// MI455X (gfx1250) — compile-verified
//
#include <hip/hip_runtime.h>
#include <stdint.h>

// Problem constants (reference: B=128, C=512, T=1024, fp32)
#define T_LEN   1024
#define TCHUNK  32                    // time steps staged per LDS tile
#define NCHUNK  (T_LEN / TCHUNK)      // 32 chunks
#define STRIDE  (TCHUNK + 4)          // 36 floats = 144B row pitch (16B aligned, <=2-way bank conflict)
#define WAVES   2                     // waves per block (wave32)
#define NBUF    4                     // quad buffer -> prefetch depth 3
#define ROWS_PER_WAVE 32              // one row (b,c) per lane
#define ROWS_PER_BLOCK (WAVES * ROWS_PER_WAVE)  // 64
#define LOADS_PER_CHUNK 8             // 8 async b128 ops/wave: 8 * (32 lanes * 16B) = 4KB = 32 rows * 128B

// Native clang vector: legal for __builtin_nontemporal_store, lowers to b128 ops.
typedef __attribute__((ext_vector_type(4))) float vf4;

__global__ __launch_bounds__(ROWS_PER_BLOCK)
void lif_scan_async_kernel(const float* __restrict__ x,
                           const float* __restrict__ beta_p,
                           const float* __restrict__ vth_p,
                           float* __restrict__ out,
                           int cmask)   // C-1 (C is a power of two: 512)
{
    // Per-wave quad-buffered tiles: 2 waves * 4 bufs * 32 rows * 36 floats * 4B = 36 KB
    __shared__ __align__(16) float tile[WAVES][NBUF][ROWS_PER_WAVE * STRIDE];

    const int tid  = threadIdx.x;
    const int wave = tid >> 5;
    const int lane = tid & 31;
    const int wave_row0 = blockIdx.x * ROWS_PER_BLOCK + wave * ROWS_PER_WAVE;

    const float beta = beta_p[0];
    const int   myrow = wave_row0 + lane;          // row this lane scans
    const float vth   = vth_p[myrow & cmask];      // per-channel threshold (row = b*C + c)

    // Async-copy lane mapping: 8 lanes cover one row-chunk (8 quads * 4 floats = 32 floats)
    const int lrow = lane >> 3;                    // 0..3 : row within group of 4
    const int q    = lane & 7;                     // 0..7 : 16-byte quad within the row chunk
    const unsigned long long xbase = (unsigned long long)(uintptr_t)x;

    // Issue one chunk's worth of async global->LDS copies for this wave (ASYNCcnt += 8)
    auto issue_loads = [&](int c, int buf) {
        const int t0 = c * TCHUNK;
#pragma unroll
        for (int j = 0; j < LOADS_PER_CHUNK; ++j) {
            const int r = 4 * j + lrow;            // row-in-wave 0..31
            unsigned goff = (unsigned)(((wave_row0 + r) * T_LEN + t0 + 4 * q) * 4);
            unsigned loff = (unsigned)(uintptr_t)&tile[wave][buf][r * STRIDE + 4 * q];
            asm volatile("global_load_async_to_lds_b128 %0, %1, %2"
                         :: "v"(loff), "v"(goff), "s"(xbase) : "memory");
        }
    };

    float mem = 0.f, spk = 0.f;

    // Prime a 3-deep pipeline: chunks 0,1,2 in flight (24 async ops)
    issue_loads(0, 0);
    issue_loads(1, 1);
    issue_loads(2, 2);

    for (int c = 0; c < NCHUNK; ++c) {
        const int buf = c & (NBUF - 1);
        const int pf  = (c + 3) & (NBUF - 1);      // buffer for chunk c+3 (== chunk c-1's buffer)

        // WAR guard: our LDS reads/writes of the buffer we are about to refill
        // (used by chunk c-1 last iteration) must retire before the async engine
        // overwrites it.
        asm volatile("s_wait_dscnt 0x0" ::: "memory");

        // Async loads retire in order among themselves: allow 8 per not-yet-needed
        // chunk beyond c, which guarantees chunk c is resident in LDS.
        if (c + 3 < NCHUNK) {
            issue_loads(c + 3, pf);
            asm volatile("s_wait_asynccnt 0x18" ::: "memory");   // <=24: c+1..c+3 pending
        } else if (c + 2 < NCHUNK) {
            asm volatile("s_wait_asynccnt 0x10" ::: "memory");   // <=16: c+1,c+2 pending
        } else if (c + 1 < NCHUNK) {
            asm volatile("s_wait_asynccnt 0x8"  ::: "memory");   // <=8 : c+1 pending
        } else {
            asm volatile("s_wait_asynccnt 0x0"  ::: "memory");
        }

        // --- LIF recurrence: lane L scans its row's TCHUNK steps from LDS,
        //     writing spikes back in place (same-wave LDS ops are in order). ---
        float* rowp = &tile[wave][buf][lane * STRIDE];
#pragma unroll
        for (int g = 0; g < TCHUNK / 4; ++g) {
            vf4 xv = *(const vf4*)(rowp + 4 * g);
            vf4 sv;
            mem = mem * beta + xv.x - spk * vth; spk = (mem > vth) ? 1.f : 0.f; sv.x = spk;
            mem = mem * beta + xv.y - spk * vth; spk = (mem > vth) ? 1.f : 0.f; sv.y = spk;
            mem = mem * beta + xv.z - spk * vth; spk = (mem > vth) ? 1.f : 0.f; sv.z = spk;
            mem = mem * beta + xv.w - spk * vth; spk = (mem > vth) ? 1.f : 0.f; sv.w = spk;
            *(vf4*)(rowp + 4 * g) = sv;
        }

        // --- Coalesced spike write-back: inverse of the load mapping.
        //     Non-temporal: output is streamed once and must not thrash L2.
        //     (Regular stores -> STOREcnt, keeping ASYNCcnt loads-only.) ---
        const int t0 = c * TCHUNK;
#pragma unroll
        for (int j = 0; j < LOADS_PER_CHUNK; ++j) {
            const int r = 4 * j + lrow;
            vf4 sv = *(const vf4*)&tile[wave][buf][r * STRIDE + 4 * q];
            __builtin_nontemporal_store(sv,
                (vf4*)(out + (size_t)((wave_row0 + r) * T_LEN + t0 + 4 * q)));
        }
    }
}

extern "C" void kernel_launch(void* const* d_in, const int* in_sizes, int n_in,
                              void* d_out, int out_size, void* d_ws, size_t ws_size,
                              hipStream_t stream) {
    const float* x    = (const float*)d_in[0];   // (B, C, T) fp32
    const float* beta = (const float*)d_in[1];   // scalar
    const float* vth  = (const float*)d_in[2];   // (C,)
    float* out = (float*)d_out;                  // (B, C, T) fp32 spikes

    const int C    = in_sizes[2];                // 512 (power of two)
    const int rows = in_sizes[0] / T_LEN;        // B*C = 65536

    dim3 grid(rows / ROWS_PER_BLOCK);            // 1024 blocks
    dim3 block(ROWS_PER_BLOCK);                  // 64 threads = 2 waves
    hipLaunchKernelGGL(lif_scan_async_kernel, grid, block, 0, stream,
                       x, beta, vth, out, C - 1);
}